// attention_decoder_2963527434662
// MI455X (gfx1250) — compile-verified
//
#include <hip/hip_runtime.h>
#include <math.h>

// MI455X / gfx1250, wave32. FP32 WMMA (16x16x4) everywhere: workload is
// memory-bound (252MB in @ 23.3TB/s ~ 11us >> 52 GFLOP on WMMA), so we keep
// exact fp32 math instead of bf16. Input staging uses the Tensor Data Mover
// (tensor_load_to_lds + s_wait_tensorcnt): every tile is uniform per
// workgroup, so one TDM descriptor replaces a global->VGPR->LDS round trip.
//
// d_in layout (setup_inputs insertion order, params flattened recursively):
//  0 current(B,9,64) 1 environment(B,1,64) 2 ts(B,20,64)
//  heads of 10 leaves each {q0w,q0b,q1w,q1b,k0w,k0b,k1w,v0w,v0b,v1w}:
//  state@3, ts@13, first@23, middle@33, last@43, out@53
//  mlp: w0@63 b0@64 w1@65 b1@66 w2@67 b2@68
// d_ws requirement: B * 2880 floats (~378 MB for B=32768).

typedef __attribute__((ext_vector_type(2))) float v2f;
typedef __attribute__((ext_vector_type(8))) float v8f;
typedef unsigned int v4u __attribute__((ext_vector_type(4)));
typedef int v4i __attribute__((ext_vector_type(4)));
typedef int v8i __attribute__((ext_vector_type(8)));

#if defined(__has_builtin)
#if __has_builtin(__builtin_amdgcn_tensor_load_to_lds) && \
    __has_builtin(__builtin_amdgcn_s_wait_tensorcnt)
#define HAVE_TDM 1
#endif
#endif

struct HeadW {
  const float *q0w, *q0b, *q1w, *q1b;
  const float *k0w, *k0b, *k1w;
  const float *v0w, *v0b, *v1w;
};

// ---------------------------------------------------------------------------
// TDM: DMA `n` contiguous f32 elements global -> LDS as a 1-D tile.
// D# group0: count=1, lds_addr, global_addr[56:32]|type=2.
// D# group1: data_size=4B; tensor_dim0=n (bits 79:48); tensor_dim1=1
// (bits 111:80); tile_dim0=n (bits 127:112); tile_dim1=1; stride0=n.
// Groups 2/3 (and the trailing group) zero (<=2-D tile, no iterate/gather).
// Caller must tdm_wait() before reading LDS.
// ---------------------------------------------------------------------------
__device__ __forceinline__ void tdm_load_1d(const float* __restrict__ gsrc,
                                            float* ldst, unsigned n) {
#ifdef HAVE_TDM
  const unsigned long long ga = (unsigned long long)(uintptr_t)gsrc;
  v4u g0;
  g0[0] = 1u;                                         // count=1 (valid), user
  g0[1] = (unsigned)(uintptr_t)ldst;                  // LDS byte offset
  g0[2] = (unsigned)ga;                               // global_addr[31:0]
  g0[3] = (unsigned)((ga >> 32) & 0x01FFFFFFu) | (2u << 30);  // [56:32]|type=2
  v8i g1;
  g1[0] = (int)(2u << 16);                 // data_size=4B, no mask/flags
  g1[1] = (int)((n & 0xFFFFu) << 16);      // tensor_dim0[15:0]
  g1[2] = (int)((n >> 16) | (1u << 16));   // tensor_dim0[31:16], tensor_dim1=1
  g1[3] = (int)((n & 0xFFFFu) << 16);      // tile_dim0=n
  g1[4] = 1;                               // tile_dim1=1, tile_dim2=0
  g1[5] = (int)n;                          // tensor_dim0_stride[31:0]
  g1[6] = 0;
  g1[7] = 0;
  v4i gz4 = {0, 0, 0, 0};
  v8i gz8 = {0, 0, 0, 0, 0, 0, 0, 0};
  // 6-arg form (clang-23 / amdgpu-toolchain signature).
  __builtin_amdgcn_tensor_load_to_lds(g0, g1, gz4, gz4, gz8, 0);
#else
  const int lane = threadIdx.x & 31;
  for (unsigned i = lane; i < n; i += 32) ldst[i] = gsrc[i];
#endif
}

__device__ __forceinline__ void tdm_wait() {
#ifdef HAVE_TDM
  __builtin_amdgcn_s_wait_tensorcnt(0);
#endif
}

// ---------------------------------------------------------------------------
// Wave-level fp32 WMMA GEMM: C(MT*16 x NT*16) = scale*(A * B) + bias, opt relu.
// A: row-major (lda), rows prepared (zero-padded) by caller.
// B: row-major KxN (ldb); if BT, B is stored NxK row-major (i.e. we use B^T).
// C: row-major (ldc). All pointers generic (LDS or global).
// ---------------------------------------------------------------------------
template <int MT, int NT, bool BT>
__device__ __forceinline__ void wave_gemm(const float* __restrict__ A, int lda,
                                          const float* __restrict__ B, int ldb,
                                          float* __restrict__ C, int ldc, int K,
                                          const float* __restrict__ bias,
                                          bool relu, float scale) {
  const int lane = threadIdx.x & 31;
  const int mrow = lane & 15;
  const int khalf = (lane >> 4) << 1;  // 0 or 2

  v8f acc[MT][NT];
#pragma unroll
  for (int i = 0; i < MT; ++i)
#pragma unroll
    for (int j = 0; j < NT; ++j)
      acc[i][j] = v8f{0.f, 0.f, 0.f, 0.f, 0.f, 0.f, 0.f, 0.f};

  for (int k0 = 0; k0 < K; k0 += 4) {
    v2f a[MT];
#pragma unroll
    for (int i = 0; i < MT; ++i) {
      const float* ap = A + (size_t)(i * 16 + mrow) * lda + (k0 + khalf);
      a[i].x = ap[0];
      a[i].y = ap[1];
    }
#pragma unroll
    for (int j = 0; j < NT; ++j) {
      v2f bfrag;
      if (BT) {
        const float* bp = B + (size_t)(j * 16 + mrow) * ldb + (k0 + khalf);
        bfrag.x = bp[0];
        bfrag.y = bp[1];
      } else {
        const float* bp = B + (size_t)(k0 + khalf) * ldb + (j * 16 + mrow);
        bfrag.x = bp[0];
        bfrag.y = bp[ldb];
      }
#pragma unroll
      for (int i = 0; i < MT; ++i)
        acc[i][j] = __builtin_amdgcn_wmma_f32_16x16x4_f32(
            false, a[i], false, bfrag, (short)0, acc[i][j], false, false);
    }
  }

#pragma unroll
  for (int i = 0; i < MT; ++i) {
    const int mbase = i * 16 + ((lane >> 4) << 3);
#pragma unroll
    for (int j = 0; j < NT; ++j) {
      const int n = j * 16 + mrow;
      const float bb = bias ? bias[n] : 0.0f;
#pragma unroll
      for (int r = 0; r < 8; ++r) {
        float v = acc[i][j][r] * scale + bb;
        if (relu) v = fmaxf(v, 0.0f);
        C[(size_t)(mbase + r) * ldc + n] = v;
      }
    }
  }
}

// Whole-wave LayerNorm over first `count` contiguous floats of LDS buffer.
__device__ __forceinline__ void wave_layernorm(float* X, int count) {
  const int lane = threadIdx.x & 31;
  float s = 0.f;
  for (int i = lane; i < count; i += 32) s += X[i];
#pragma unroll
  for (int off = 16; off >= 1; off >>= 1) s += __shfl_xor(s, off, 32);
  const float mu = s / (float)count;
  float v = 0.f;
  for (int i = lane; i < count; i += 32) {
    float d = X[i] - mu;
    v += d * d;
  }
#pragma unroll
  for (int off = 16; off >= 1; off >>= 1) v += __shfl_xor(v, off, 32);
  const float rstd = rsqrtf(v / (float)count + 1e-5f);
  for (int i = lane; i < count; i += 32) X[i] = (X[i] - mu) * rstd;
}

// Row-serial softmax over LDS scores; zeroes padded columns [cols, colpad).
__device__ __forceinline__ void softmax_rows(float* S, int lds, int rows,
                                             int cols, int colpad) {
  const int lane = threadIdx.x & 31;
  for (int r = lane; r < rows; r += 32) {
    float* row = S + (size_t)r * lds;
    float m = -3.4e38f;
    for (int c = 0; c < cols; ++c) m = fmaxf(m, row[c]);
    float sum = 0.f;
    for (int c = 0; c < cols; ++c) {
      float e = expf(row[c] - m);
      row[c] = e;
      sum += e;
    }
    const float inv = 1.0f / sum;
    for (int c = 0; c < cols; ++c) row[c] *= inv;
    for (int c = cols; c < colpad; ++c) row[c] = 0.f;
  }
}

#define SCALE32 0.17677669529663689f  // 1/sqrt(32)
#define SCALE8 0.35355339059327373f   // 1/sqrt(8)

// ---------------- state head: LN + qkv + self-attn; emits q_s and seq[0:10] --
__global__ __launch_bounds__(32) void k_state(const float* __restrict__ cur,
                                              const float* __restrict__ env,
                                              HeadW w,
                                              float* __restrict__ qs_out,
                                              float* __restrict__ seq) {
  __shared__ __align__(16) float X[16 * 64];
  __shared__ __align__(16) float H[16 * 32];
  __shared__ __align__(16) float Q[16 * 32];
  __shared__ __align__(16) float Km[16 * 32];
  __shared__ __align__(16) float V[16 * 32];
  __shared__ __align__(16) float S[16 * 16];
  const int b = blockIdx.x, lane = threadIdx.x;

  tdm_load_1d(cur + (size_t)b * 9 * 64, X, 9 * 64);       // rows 0-8
  tdm_load_1d(env + (size_t)b * 64, X + 9 * 64, 64);      // row 9
  for (int i = lane; i < 6 * 64; i += 32) X[10 * 64 + i] = 0.f;
  tdm_wait();
  wave_layernorm(X, 10 * 64);

  wave_gemm<1, 2, false>(X, 64, w.q0w, 32, H, 32, 64, w.q0b, true, 1.f);
  wave_gemm<1, 2, false>(H, 32, w.q1w, 32, Q, 32, 32, w.q1b, false, 1.f);
  wave_gemm<1, 2, false>(X, 64, w.k0w, 32, H, 32, 64, w.k0b, true, 1.f);
  wave_gemm<1, 2, false>(H, 32, w.k1w, 32, Km, 32, 32, nullptr, false, 1.f);
  wave_gemm<1, 2, false>(X, 64, w.v0w, 32, H, 32, 64, w.v0b, true, 1.f);
  wave_gemm<1, 2, false>(H, 32, w.v1w, 32, V, 32, 32, nullptr, false, 1.f);

  wave_gemm<1, 1, true>(Q, 32, Km, 32, S, 16, 32, nullptr, false, SCALE32);
  softmax_rows(S, 16, 10, 10, 16);
  wave_gemm<1, 2, false>(S, 16, V, 32, H, 32, 16, nullptr, false, 1.f);

  float* qd = qs_out + (size_t)b * 10 * 32;
  for (int i = lane; i < 10 * 32; i += 32) qd[i] = Q[i];
  float* sd = seq + (size_t)b * 60 * 32;
  for (int i = lane; i < 10 * 32; i += 32) sd[i] = H[i];
}

// ---- windowed ts heads (first/middle/last): kv-only + cross-attn vs q_s -----
__global__ __launch_bounds__(32) void k_win(const float* __restrict__ ts,
                                            int off, HeadW w,
                                            const float* __restrict__ qs,
                                            float* __restrict__ seq, int slot) {
  __shared__ __align__(16) float X[16 * 64];
  __shared__ __align__(16) float H[16 * 32];
  __shared__ __align__(16) float Q[16 * 32];
  __shared__ __align__(16) float Km[16 * 32];
  __shared__ __align__(16) float V[16 * 32];
  __shared__ __align__(16) float S[16 * 16];
  const int b = blockIdx.x, lane = threadIdx.x;

  tdm_load_1d(ts + ((size_t)b * 20 + off) * 64, X, 10 * 64);
  tdm_load_1d(qs + (size_t)b * 10 * 32, Q, 10 * 32);  // overlaps the X load; done before LN
  for (int i = lane; i < 6 * 64; i += 32) X[10 * 64 + i] = 0.f;
  for (int i = lane; i < 6 * 32; i += 32) Q[10 * 32 + i] = 0.f;
  tdm_wait();
  wave_layernorm(X, 10 * 64);

  wave_gemm<1, 2, false>(X, 64, w.k0w, 32, H, 32, 64, w.k0b, true, 1.f);
  wave_gemm<1, 2, false>(H, 32, w.k1w, 32, Km, 32, 32, nullptr, false, 1.f);
  wave_gemm<1, 2, false>(X, 64, w.v0w, 32, H, 32, 64, w.v0b, true, 1.f);
  wave_gemm<1, 2, false>(H, 32, w.v1w, 32, V, 32, 32, nullptr, false, 1.f);

  wave_gemm<1, 1, true>(Q, 32, Km, 32, S, 16, 32, nullptr, false, SCALE32);
  softmax_rows(S, 16, 10, 10, 16);
  wave_gemm<1, 2, false>(S, 16, V, 32, H, 32, 16, nullptr, false, 1.f);

  float* sd = seq + (size_t)b * 60 * 32 + (size_t)slot * 10 * 32;
  for (int i = lane; i < 10 * 32; i += 32) sd[i] = H[i];
}

// ---------------- full ts head (S=20 -> padded 32): self-attn ----------------
__global__ __launch_bounds__(32) void k_ts(const float* __restrict__ ts,
                                           HeadW w, float* __restrict__ seq) {
  __shared__ __align__(16) float X[32 * 64];
  __shared__ __align__(16) float H[32 * 32];
  __shared__ __align__(16) float Q[32 * 32];
  __shared__ __align__(16) float Km[32 * 32];
  __shared__ __align__(16) float V[32 * 32];
  __shared__ __align__(16) float S[32 * 32];
  const int b = blockIdx.x, lane = threadIdx.x;

  tdm_load_1d(ts + (size_t)b * 20 * 64, X, 20 * 64);
  for (int i = lane; i < 12 * 64; i += 32) X[20 * 64 + i] = 0.f;
  tdm_wait();
  wave_layernorm(X, 20 * 64);

  wave_gemm<2, 2, false>(X, 64, w.q0w, 32, H, 32, 64, w.q0b, true, 1.f);
  wave_gemm<2, 2, false>(H, 32, w.q1w, 32, Q, 32, 32, w.q1b, false, 1.f);
  wave_gemm<2, 2, false>(X, 64, w.k0w, 32, H, 32, 64, w.k0b, true, 1.f);
  wave_gemm<2, 2, false>(H, 32, w.k1w, 32, Km, 32, 32, nullptr, false, 1.f);
  wave_gemm<2, 2, false>(X, 64, w.v0w, 32, H, 32, 64, w.v0b, true, 1.f);
  wave_gemm<2, 2, false>(H, 32, w.v1w, 32, V, 32, 32, nullptr, false, 1.f);

  wave_gemm<2, 2, true>(Q, 32, Km, 32, S, 32, 32, nullptr, false, SCALE32);
  softmax_rows(S, 32, 20, 20, 32);
  wave_gemm<2, 2, false>(S, 32, V, 32, H, 32, 32, nullptr, false, 1.f);

  float* sd = seq + (size_t)b * 60 * 32 + 40 * 32;
  for (int i = lane; i < 20 * 32; i += 32) sd[i] = H[i];
}

// -------- out head: LN(seq 60x32) + qkv(32->16->8) + 60x60 attn + LN --------
__global__ __launch_bounds__(32) void k_out(const float* __restrict__ seq,
                                            HeadW w, float* __restrict__ att) {
  __shared__ __align__(16) float X[64 * 32];
  __shared__ __align__(16) float H[64 * 16];
  __shared__ __align__(16) float Q[64 * 16];
  __shared__ __align__(16) float Km[64 * 16];
  __shared__ __align__(16) float V[64 * 16];
  __shared__ __align__(16) float S[64 * 64];
  __shared__ float W2q[16 * 16], W2k[16 * 16], W2v[16 * 16], B2q[16];
  const int b = blockIdx.x, lane = threadIdx.x;

  tdm_load_1d(seq + (size_t)b * 60 * 32, X, 60 * 32);
  for (int i = lane; i < 4 * 32; i += 32) X[60 * 32 + i] = 0.f;

  // stage layer-2 weights zero-padded 16x8 -> 16x16 (overlaps the TDM)
  for (int i = lane; i < 16 * 16; i += 32) {
    const int r = i >> 4, c = i & 15;
    W2q[i] = (c < 8) ? w.q1w[r * 8 + c] : 0.f;
    W2k[i] = (c < 8) ? w.k1w[r * 8 + c] : 0.f;
    W2v[i] = (c < 8) ? w.v1w[r * 8 + c] : 0.f;
  }
  if (lane < 16) B2q[lane] = (lane < 8) ? w.q1b[lane] : 0.f;

  tdm_wait();
  wave_layernorm(X, 60 * 32);

  wave_gemm<4, 1, false>(X, 32, w.q0w, 16, H, 16, 32, w.q0b, true, 1.f);
  wave_gemm<4, 1, false>(H, 16, W2q, 16, Q, 16, 16, B2q, false, 1.f);
  wave_gemm<4, 1, false>(X, 32, w.k0w, 16, H, 16, 32, w.k0b, true, 1.f);
  wave_gemm<4, 1, false>(H, 16, W2k, 16, Km, 16, 16, nullptr, false, 1.f);
  wave_gemm<4, 1, false>(X, 32, w.v0w, 16, H, 16, 32, w.v0b, true, 1.f);
  wave_gemm<4, 1, false>(H, 16, W2v, 16, V, 16, 16, nullptr, false, 1.f);

  wave_gemm<4, 4, true>(Q, 16, Km, 16, S, 64, 16, nullptr, false, SCALE8);
  softmax_rows(S, 64, 60, 60, 64);
  wave_gemm<4, 1, false>(S, 64, V, 16, H, 16, 64, nullptr, false, 1.f);

  // compact 60x8 -> contiguous 480, final LN, write out
  for (int i = lane; i < 480; i += 32) {
    const int r = i >> 3, c = i & 7;
    X[i] = H[r * 16 + c];
  }
  wave_layernorm(X, 480);
  float* ad = att + (size_t)b * 480;
  for (int i = lane; i < 480; i += 32) ad[i] = X[i];
}

// -------- batch-major MLP GEMMs: one wave per 16 batch rows -----------------
// A tile staged to LDS by TDM (contiguous 30KB DMA beats strided VGPR loads).
__global__ __launch_bounds__(32) void k_mlp1(const float* __restrict__ att,
                                             const float* __restrict__ w,
                                             const float* __restrict__ bias,
                                             float* __restrict__ h1) {
  __shared__ __align__(16) float Xa[16 * 480];
  const size_t m0 = (size_t)blockIdx.x * 16;
  tdm_load_1d(att + m0 * 480, Xa, 16 * 480);
  tdm_wait();
  wave_gemm<1, 8, false>(Xa, 480, w, 128, h1 + m0 * 128, 128, 480, bias, true,
                         1.f);
}

__global__ __launch_bounds__(32) void k_mlp2(const float* __restrict__ h1,
                                             const float* __restrict__ w,
                                             const float* __restrict__ bias,
                                             float* __restrict__ h2) {
  __shared__ __align__(16) float Xa[16 * 128];
  const size_t m0 = (size_t)blockIdx.x * 16;
  tdm_load_1d(h1 + m0 * 128, Xa, 16 * 128);
  tdm_wait();
  wave_gemm<1, 2, false>(Xa, 128, w, 32, h2 + m0 * 32, 32, 128, bias, true,
                         1.f);
}

// -------- final 32->2 head: mu and softplus(sigma)+1e-6 ---------------------
__global__ __launch_bounds__(256) void k_head(const float* __restrict__ h2,
                                              const float* __restrict__ w,
                                              const float* __restrict__ bias,
                                              float* __restrict__ out, int B) {
  const int i = blockIdx.x * 256 + threadIdx.x;
  if (i >= B) return;
  const float* x = h2 + (size_t)i * 32;
  float a0 = bias[0], a1 = bias[1];
#pragma unroll
  for (int k = 0; k < 32; ++k) {
    const float xv = x[k];
    a0 += xv * w[k * 2 + 0];
    a1 += xv * w[k * 2 + 1];
  }
  out[i] = a0;  // mu
  const float sp = fmaxf(a1, 0.f) + log1pf(expf(-fabsf(a1)));  // softplus
  out[(size_t)B + i] = sp + 1e-6f;  // sigma
}

static HeadW mkhead(void* const* d, int i) {
  HeadW h;
  h.q0w = (const float*)d[i + 0];
  h.q0b = (const float*)d[i + 1];
  h.q1w = (const float*)d[i + 2];
  h.q1b = (const float*)d[i + 3];
  h.k0w = (const float*)d[i + 4];
  h.k0b = (const float*)d[i + 5];
  h.k1w = (const float*)d[i + 6];
  h.v0w = (const float*)d[i + 7];
  h.v0b = (const float*)d[i + 8];
  h.v1w = (const float*)d[i + 9];
  return h;
}

extern "C" void kernel_launch(void* const* d_in, const int* in_sizes, int n_in,
                              void* d_out, int out_size, void* d_ws,
                              size_t ws_size, hipStream_t stream) {
  const float* cur = (const float*)d_in[0];
  const float* env = (const float*)d_in[1];
  const float* ts = (const float*)d_in[2];
  const int B = in_sizes[0] / (9 * 64);  // 32768; multiple of 16

  // workspace: B*2880 floats
  float* ws = (float*)d_ws;
  float* qs = ws;                       // B*320
  float* seq = qs + (size_t)B * 320;    // B*1920
  float* att = seq + (size_t)B * 1920;  // B*480
  float* h1 = att + (size_t)B * 480;    // B*128
  float* h2 = h1 + (size_t)B * 128;     // B*32

  const float* m0w = (const float*)d_in[63];
  const float* m0b = (const float*)d_in[64];
  const float* m1w = (const float*)d_in[65];
  const float* m1b = (const float*)d_in[66];
  const float* m2w = (const float*)d_in[67];
  const float* m2b = (const float*)d_in[68];

  k_state<<<B, 32, 0, stream>>>(cur, env, mkhead(d_in, 3), qs, seq);
  k_win<<<B, 32, 0, stream>>>(ts, 0, mkhead(d_in, 23), qs, seq, 1);
  k_win<<<B, 32, 0, stream>>>(ts, 5, mkhead(d_in, 33), qs, seq, 2);
  k_win<<<B, 32, 0, stream>>>(ts, 10, mkhead(d_in, 43), qs, seq, 3);
  k_ts<<<B, 32, 0, stream>>>(ts, mkhead(d_in, 13), seq);
  k_out<<<B, 32, 0, stream>>>(seq, mkhead(d_in, 53), att);
  k_mlp1<<<B / 16, 32, 0, stream>>>(att, m0w, m0b, h1);
  k_mlp2<<<B / 16, 32, 0, stream>>>(h1, m1w, m1b, h2);
  k_head<<<(B + 255) / 256, 256, 0, stream>>>(h2, m2w, m2b, (float*)d_out, B);
}